// EncoderRNN_71012989272777
// MI455X (gfx1250) — compile-verified
//
#include <hip/hip_runtime.h>
#include <hip/hip_bf16.h>
#include <stdint.h>

#define CH      1024
#define GCH     4096           // 4*CH
#define LATN    128
#define EOS_TOK 1
#define SCAN_BLOCKS  128
#define SCAN_THREADS 256
#define ROWS_PER_BLOCK (GCH / SCAN_BLOCKS)   // 32 gate rows per block

typedef __attribute__((ext_vector_type(2))) float v2f;
typedef __attribute__((ext_vector_type(8))) float v8f;
typedef __attribute__((ext_vector_type(4))) unsigned int u32x4;
typedef __attribute__((ext_vector_type(4))) int i32x4;
typedef __attribute__((ext_vector_type(8))) int i32x8;

__device__ __forceinline__ float sigmoidf_(float x) {
    return 1.0f / (1.0f + expf(-x));
}

// ---------------------------------------------------------------------------
// Kernel 0: first-EOS index (argmax semantics: 0 if none), init h/c, reset
// grid-barrier state (re-done every launch -> graph-replay deterministic).
// ---------------------------------------------------------------------------
__global__ void __launch_bounds__(1024)
prep_kernel(const int* __restrict__ tok, const float* __restrict__ h0,
            const float* __restrict__ c0, float* __restrict__ h,
            float* __restrict__ c, int* __restrict__ hdr, int T) {
    __shared__ int smin;
    if (threadIdx.x == 0) smin = 0x7FFFFFFF;
    __syncthreads();
    for (int i = threadIdx.x; i < T; i += blockDim.x)
        if (tok[i] == EOS_TOK) atomicMin(&smin, i);
    __syncthreads();
    if (threadIdx.x == 0) {
        hdr[0] = (smin == 0x7FFFFFFF) ? 0 : smin;   // steps to run
        hdr[1] = 0;                                  // barrier counter
        hdr[2] = 0;                                  // barrier generation
    }
    for (int i = threadIdx.x; i < CH; i += blockDim.x) {
        h[i] = h0[i];
        c[i] = c0[i];
    }
}

// ---------------------------------------------------------------------------
// Kernel 1: XW[t, :] = emb[tok[t]] @ W_ih^T + (b_ih + b_hh)
// fp32 GEMM on the matrix pipe: V_WMMA_F32_16X16X4_F32, one wave per tile.
// ---------------------------------------------------------------------------
__global__ void __launch_bounds__(256)
xw_wmma_kernel(const int* __restrict__ tok, const float* __restrict__ emb,
               const float* __restrict__ Wih, const float* __restrict__ bih,
               const float* __restrict__ bhh, float* __restrict__ XW) {
    const int lane = threadIdx.x & 31;
    const int wave = blockIdx.x * (blockDim.x >> 5) + (threadIdx.x >> 5);
    const int ntn  = GCH / 16;                 // 256 tiles along N
    const int mt = wave / ntn, nt = wave % ntn;
    const int m0 = mt * 16, n0 = nt * 16;
    const int l15 = lane & 15;
    const int kk  = (lane >> 4) << 1;          // 0 or 2

    const int    tokrow = tok[m0 + l15];
    const float* aptr   = emb + (size_t)tokrow * CH;          // A row (M=l15)
    const float* bptr   = Wih + (size_t)(n0 + l15) * CH;      // B col (N=l15)

    v8f acc = {};
    for (int k = 0; k < CH; k += 4) {
        v2f a, b;
        a.x = aptr[k + kk];
        a.y = aptr[k + kk + 1];
        b.x = bptr[k + kk];
        b.y = bptr[k + kk + 1];
        acc = __builtin_amdgcn_wmma_f32_16x16x4_f32(
            false, a, false, b, (short)0, acc, false, false);
    }

    const int   crow = (lane >> 4) << 3;       // 0 or 8
    const float bias = bih[n0 + l15] + bhh[n0 + l15];
#pragma unroll
    for (int r = 0; r < 8; ++r)
        XW[(size_t)(m0 + crow + r) * GCH + (n0 + l15)] = acc[r] + bias;
}

// ---------------------------------------------------------------------------
// Grid-wide sense-reversing barrier (atomic counter + generation).
// ---------------------------------------------------------------------------
__device__ __forceinline__ void grid_sync(int* cnt, volatile int* gen, int nb) {
    __threadfence();            // each thread publishes its own prior stores
    __syncthreads();
    if (threadIdx.x == 0) {
        int g = *gen;
        if (atomicAdd(cnt, 1) == nb - 1) {
            *cnt = 0;
            __threadfence();
            atomicAdd((int*)gen, 1);
        } else {
            while (*gen == g) __builtin_amdgcn_s_sleep(1);
        }
    }
    __syncthreads();
}

// ---------------------------------------------------------------------------
// Kernel 2: sequential LSTM scan, W_hh slice resident in LDS.
// 128 blocks x 256 threads. Block b owns gate rows [b*32, b*32+32): its
// 128 KB W_hh slice is DMA'd once into LDS by the Tensor Data Mover and
// reused for all ~1023 steps (LDS-only GEMV instead of 16 GB L2 streaming).
// Two grid barriers per step around the h/c nonlinearity.
// ---------------------------------------------------------------------------
__global__ void __launch_bounds__(SCAN_THREADS)
lstm_scan_kernel(const float* __restrict__ Whh, const float* __restrict__ XW,
                 float* __restrict__ h, float* __restrict__ c,
                 float* __restrict__ gates, int* __restrict__ hdr) {
    __shared__ float ws[ROWS_PER_BLOCK * CH];   // 128 KB W_hh slice
    __shared__ float hs[CH];                    // 4 KB h broadcast
    const int tid   = threadIdx.x;
    const int lane  = tid & 31;
    const int wv    = tid >> 5;                 // wave in block: 0..7
    const int gtid  = blockIdx.x * SCAN_THREADS + tid;
    const int rowbase = blockIdx.x * ROWS_PER_BLOCK;
    int*          cnt = hdr + 1;
    volatile int* gen = hdr + 2;
    const int steps = hdr[0];

    // ---- TDM fill: 2-D tile (1024 x 32 f32) from Whh[rowbase..] into ws ----
    if (wv == 0) {
        const uint64_t ga  = (uint64_t)(uintptr_t)(Whh + (size_t)rowbase * CH);
        const uint32_t lds = (uint32_t)(uintptr_t)(void*)ws;  // LDS byte offset
        u32x4 g0;
        g0[0] = 1u;                                   // count=1 user descriptor
        g0[1] = lds;                                  // lds_addr
        g0[2] = (uint32_t)ga;                         // global_addr[31:0]
        g0[3] = (uint32_t)((ga >> 32) & 0x1FFFFFFu)   // global_addr[56:32]
              | (2u << 30);                           // type=2 ("image")
        i32x8 g1;
        g1[0] = 2 << 16;                              // data_size=4B; wg_mask=0
        g1[1] = (CH & 0xFFFF) << 16;                  // tensor_dim0 = 1024
        g1[2] = (GCH & 0xFFFF) << 16;                 // tensor_dim1 = 4096
        g1[3] = (CH & 0xFFFF) << 16;                  // tile_dim0 = 1024
        g1[4] = ROWS_PER_BLOCK;                       // tile_dim1 = 32
        g1[5] = CH;                                   // tensor_dim0_stride lo
        g1[6] = 0;                                    // stride hi / dim1_stride lo
        g1[7] = 0;
        i32x4 gz4 = {};                               // 2-D: groups 2/3 unused
        i32x8 gz8 = {};
        __builtin_amdgcn_tensor_load_to_lds(g0, g1, gz4, gz4, gz8, 0);
        __builtin_amdgcn_s_wait_tensorcnt(0);
    }
    __syncthreads();

    const float* wr0 = ws + (wv * 4 + 0) * CH;
    const float* wr1 = ws + (wv * 4 + 1) * CH;
    const float* wr2 = ws + (wv * 4 + 2) * CH;
    const float* wr3 = ws + (wv * 4 + 3) * CH;

    for (int t = 0; t < steps; ++t) {
        for (int i = tid; i < CH; i += SCAN_THREADS) hs[i] = h[i];
        __syncthreads();

        float s0 = 0.f, s1 = 0.f, s2 = 0.f, s3 = 0.f;
#pragma unroll 4
        for (int k = lane; k < CH; k += 32) {
            const float hv = hs[k];
            s0 = fmaf(hv, wr0[k], s0);
            s1 = fmaf(hv, wr1[k], s1);
            s2 = fmaf(hv, wr2[k], s2);
            s3 = fmaf(hv, wr3[k], s3);
        }
        for (int off = 16; off; off >>= 1) {
            s0 += __shfl_xor(s0, off, 32);
            s1 += __shfl_xor(s1, off, 32);
            s2 += __shfl_xor(s2, off, 32);
            s3 += __shfl_xor(s3, off, 32);
        }
        if (lane == 0) {
            const float* xw = XW + (size_t)t * GCH;
            const int j = rowbase + wv * 4;
            gates[j + 0] = s0 + xw[j + 0];
            gates[j + 1] = s1 + xw[j + 1];
            gates[j + 2] = s2 + xw[j + 2];
            gates[j + 3] = s3 + xw[j + 3];
        }

        grid_sync(cnt, gen, gridDim.x);

        if (gtid < CH) {
            const float ig = sigmoidf_(gates[gtid]);
            const float fg = sigmoidf_(gates[CH + gtid]);
            const float gg = tanhf(gates[2 * CH + gtid]);
            const float og = sigmoidf_(gates[3 * CH + gtid]);
            const float cn = fg * c[gtid] + ig * gg;
            c[gtid] = cn;
            h[gtid] = og * tanhf(cn);
        }

        grid_sync(cnt, gen, gridDim.x);
    }
}

// ---------------------------------------------------------------------------
// Kernel 3: VAE head.  mean/logv = [h,c] @ W^T + b ; z = eps*exp(.5*logv)+mean
// out = concat(z, mean, logv)
// ---------------------------------------------------------------------------
__global__ void __launch_bounds__(256)
head_kernel(const float* __restrict__ h, const float* __restrict__ c,
            const float* __restrict__ Wm, const float* __restrict__ bm,
            const float* __restrict__ Wl, const float* __restrict__ bl,
            const float* __restrict__ eps, float* __restrict__ out) {
    __shared__ float sm[LATN], sl[LATN];
    const int tid = threadIdx.x;
    const int i   = tid & (LATN - 1);
    const bool isMean = (tid < LATN);
    const float* row = (isMean ? Wm : Wl) + (size_t)i * (2 * CH);
    float s = isMean ? bm[i] : bl[i];
    for (int k = 0; k < CH; ++k) s = fmaf(h[k], row[k], s);
    for (int k = 0; k < CH; ++k) s = fmaf(c[k], row[CH + k], s);
    if (isMean) sm[i] = s; else sl[i] = s;
    __syncthreads();
    if (isMean) {
        const float mean = sm[i], logv = sl[i];
        out[i]            = eps[i] * expf(0.5f * logv) + mean;
        out[LATN + i]     = mean;
        out[2 * LATN + i] = logv;
    }
}

// ---------------------------------------------------------------------------
extern "C" void kernel_launch(void* const* d_in, const int* in_sizes, int n_in,
                              void* d_out, int out_size, void* d_ws, size_t ws_size,
                              hipStream_t stream) {
    const int*   tok = (const int*)  d_in[0];
    const float* h0  = (const float*)d_in[1];
    const float* c0  = (const float*)d_in[2];
    const float* eps = (const float*)d_in[3];
    const float* emb = (const float*)d_in[4];
    const float* Wih = (const float*)d_in[5];
    const float* Whh = (const float*)d_in[6];
    const float* bih = (const float*)d_in[7];
    const float* bhh = (const float*)d_in[8];
    const float* Wm  = (const float*)d_in[9];
    const float* bm  = (const float*)d_in[10];
    const float* Wl  = (const float*)d_in[11];
    const float* bl  = (const float*)d_in[12];
    const int T = in_sizes[0];              // 1024

    // workspace layout: [256B header][XW: T*GCH f32][h: CH][c: CH][gates: GCH]
    int*   hdr   = (int*)d_ws;
    float* XW    = (float*)((char*)d_ws + 256);
    float* h     = XW + (size_t)T * GCH;
    float* c     = h + CH;
    float* gates = c + CH;

    prep_kernel<<<1, 1024, 0, stream>>>(tok, h0, c0, h, c, hdr, T);

    const int tiles = (T / 16) * (GCH / 16);    // one wave per 16x16 tile
    xw_wmma_kernel<<<tiles / 8, 256, 0, stream>>>(tok, emb, Wih, bih, bhh, XW);

    lstm_scan_kernel<<<SCAN_BLOCKS, SCAN_THREADS, 0, stream>>>(Whh, XW, h, c,
                                                               gates, hdr);

    head_kernel<<<1, 256, 0, stream>>>(h, c, Wm, bm, Wl, bl, eps, (float*)d_out);
}